// PureALU_62380105007302
// MI455X (gfx1250) — compile-verified
//
#include <hip/hip_runtime.h>

// CDNA5 / gfx1250: f32 WMMA chain, 2 tiles/wave, async global->LDS weight
// staging in 2-block epochs (double-buffered, fully unrolled), bias folded
// into WMMA via a ones-column A-fragment. Exact f32 arithmetic is required by
// the reference's S=100 step-function bit tricks -> V_WMMA_F32_16X16X4_F32.
typedef __attribute__((ext_vector_type(2))) float v2f;  // A/B frag (2 VGPRs)
typedef __attribute__((ext_vector_type(8))) float v8f;  // 16x16 f32 C/D (8 VGPRs)

#define WAVES_PER_WG 8
#define THREADS (WAVES_PER_WG * 32)
#define TPW 2                      // tiles per wave
#define SB  2                      // blocks staged per async epoch

__device__ __forceinline__ void async_cp16(const float* g, const float* l) {
    // Low 32 bits of a generic shared-aperture address are the LDS byte offset.
    unsigned lds_off = (unsigned)(unsigned long long)l;
    asm volatile("global_load_async_to_lds_b128 %0, %1, off"
                 :: "v"(lds_off), "v"(g) : "memory");
}
__device__ __forceinline__ void wait_async0() {
    asm volatile("s_wait_asynccnt 0x0" ::: "memory");
}

// Layouts (ISA 7.12.2, 32-bit):
//   C/D 16x16 : VGPR r, lanes 0-15 -> M=r, lanes 16-31 -> M=r+8, N = lane%16
//   A   16x4  : lane L -> M=L%16 ; VGPR a -> K = a + 2*(L/16)
//   B   4x16  : lane L -> N=L%16 ; VGPR b -> K = b + 2*(L/16)
__global__ __launch_bounds__(THREADS) void palu_chain_kernel(
    const float* __restrict__ x,
    const float* __restrict__ Wu, const float* __restrict__ bu,
    const float* __restrict__ Wg, const float* __restrict__ bg,
    const float* __restrict__ Wd,
    float* __restrict__ out, int n_rows, int n_blk)
{
    // Per-wave bounce slabs (same-wave LDS is in-order -> no barriers needed).
    __shared__ float Hs[WAVES_PER_WG * TPW][16 * 16];   // h tiles, row-major
    __shared__ float Ps[WAVES_PER_WG * TPW][16 * 32];   // relu(up)*gate
    // Workgroup-shared, double-buffered 2-block weight groups (async DMA).
    __shared__ float WuS[2][SB * 512];
    __shared__ float WgS[2][SB * 512];
    __shared__ float WdS[2][SB * 512];
    __shared__ float bS [2][SB * 64];                   // per blk: bu[32]++bg[32]

    const int tid  = threadIdx.x;
    const int lane = tid & 31;
    const int wave = tid >> 5;
    const int hi   = lane >> 4;
    const int lo   = lane & 15;

    // Clamp instead of early-return so every wave reaches every barrier.
    const long max_tile = (long)(n_rows / 16) - 1;
    long t0 = ((long)blockIdx.x * WAVES_PER_WG + wave) * TPW;
    long t1 = t0 + 1;
    if (t0 > max_tile) t0 = max_tile;
    if (t1 > max_tile) t1 = max_tile;
    const long row0 = t0 * 16, row1 = t1 * 16;

    float* hs0 = Hs[wave * TPW + 0];
    float* hs1 = Hs[wave * TPW + 1];
    float* ps0 = Ps[wave * TPW + 0];
    float* ps1 = Ps[wave * TPW + 1];

    // Cooperative async stage of an epoch (SB consecutive blocks) of weights.
    // Weights are contiguous across blocks -> one linear b128 sweep per array.
    auto stage_group = [&](int gbuf, int base) {
        const int nb  = (n_blk - base < SB) ? (n_blk - base) : SB;
        const int nch = nb * 128;                       // b128 chunks per array
        const float* wu = Wu + (size_t)base * 512;
        const float* wg = Wg + (size_t)base * 512;
        const float* wd = Wd + (size_t)base * 512;
        if (tid < nch) {                                // nch is 128 or 256
            async_cp16(wu + tid * 4, &WuS[gbuf][tid * 4]);
            async_cp16(wg + tid * 4, &WgS[gbuf][tid * 4]);
            async_cp16(wd + tid * 4, &WdS[gbuf][tid * 4]);
        }
        const int nbch = nb * 8;                        // bias b128 chunks
        if (tid < nbch) {
            async_cp16(bu + (size_t)base * 32 + tid * 4,
                       &bS[gbuf][(tid >> 3) * 64 + (tid & 7) * 4]);
        } else if (tid < 2 * nbch) {
            const int t = tid - nbch;
            async_cp16(bg + (size_t)base * 32 + t * 4,
                       &bS[gbuf][(t >> 3) * 64 + 32 + (t & 7) * 4]);
        }
    };

    // Ones-column A fragment: A[m][k'] = (k'==0);  k' = a + 2*hi.
    v2f aOne;
    aOne.x = (hi == 0) ? 1.0f : 0.0f;
    aOne.y = 0.0f;

    // One FFN block for both tiles; slab pointers carry compile-time offsets.
    auto do_block = [&](const float* wuS, const float* wgS,
                        const float* wdS, const float* bSj,
                        v8f& h0, v8f& h1) {
        // Bias B-fragments (only row k'=0 is multiplied by nonzero A).
        const float vbu0 = bSj[lo],      vbu1 = bSj[16 + lo];
        const float vbg0 = bSj[32 + lo], vbg1 = bSj[48 + lo];
        v2f bbu0 = {vbu0, vbu0}, bbu1 = {vbu1, vbu1};
        v2f bbg0 = {vbg0, vbg0}, bbg1 = {vbg1, vbg1};

        // Seed accumulators with bias via one WMMA each (C = inline 0);
        // each seed value feeds BOTH tiles' chains (no copies).
        v8f z = {};
        const v8f s_u0 = __builtin_amdgcn_wmma_f32_16x16x4_f32(false, aOne, false, bbu0, (short)0, z, false, false);
        const v8f s_u1 = __builtin_amdgcn_wmma_f32_16x16x4_f32(false, aOne, false, bbu1, (short)0, z, false, false);
        const v8f s_g0 = __builtin_amdgcn_wmma_f32_16x16x4_f32(false, aOne, false, bbg0, (short)0, z, false, false);
        const v8f s_g1 = __builtin_amdgcn_wmma_f32_16x16x4_f32(false, aOne, false, bbg1, (short)0, z, false, false);

        // A-fragments of both H tiles (K chunks of 4).
        v2f aH0[4], aH1[4];
        #pragma unroll
        for (int kk = 0; kk < 4; ++kk) {
            aH0[kk] = *(const v2f*)&hs0[lo * 16 + 4 * kk + 2 * hi];
            aH1[kk] = *(const v2f*)&hs1[lo * 16 + 4 * kk + 2 * hi];
        }

        v8f u0a = s_u0, u1a = s_u1, g0a = s_g0, g1a = s_g1;   // tile 0
        v8f u0b = s_u0, u1b = s_u1, g0b = s_g0, g1b = s_g1;   // tile 1

        // up = h@Wu^T + bu, gate = h@Wg^T + bg; weight frags shared by tiles.
        #pragma unroll
        for (int kk = 0; kk < 4; ++kk) {
            const int f = 4 * kk + 2 * hi;
            const v2f b_u0 = *(const v2f*)&wuS[(lo)      * 16 + f];
            const v2f b_u1 = *(const v2f*)&wuS[(16 + lo) * 16 + f];
            const v2f b_g0 = *(const v2f*)&wgS[(lo)      * 16 + f];
            const v2f b_g1 = *(const v2f*)&wgS[(16 + lo) * 16 + f];
            u0a = __builtin_amdgcn_wmma_f32_16x16x4_f32(false, aH0[kk], false, b_u0, (short)0, u0a, false, false);
            u0b = __builtin_amdgcn_wmma_f32_16x16x4_f32(false, aH1[kk], false, b_u0, (short)0, u0b, false, false);
            u1a = __builtin_amdgcn_wmma_f32_16x16x4_f32(false, aH0[kk], false, b_u1, (short)0, u1a, false, false);
            u1b = __builtin_amdgcn_wmma_f32_16x16x4_f32(false, aH1[kk], false, b_u1, (short)0, u1b, false, false);
            g0a = __builtin_amdgcn_wmma_f32_16x16x4_f32(false, aH0[kk], false, b_g0, (short)0, g0a, false, false);
            g0b = __builtin_amdgcn_wmma_f32_16x16x4_f32(false, aH1[kk], false, b_g0, (short)0, g0b, false, false);
            g1a = __builtin_amdgcn_wmma_f32_16x16x4_f32(false, aH0[kk], false, b_g1, (short)0, g1a, false, false);
            g1b = __builtin_amdgcn_wmma_f32_16x16x4_f32(false, aH1[kk], false, b_g1, (short)0, g1b, false, false);
        }

        // p = relu(up) * gate, bounced through per-wave slabs (C/D -> A).
        #pragma unroll
        for (int r = 0; r < 8; ++r) {
            const int m = r + 8 * hi;
            ps0[m * 32 + lo]      = fmaxf(u0a[r], 0.0f) * g0a[r];
            ps0[m * 32 + 16 + lo] = fmaxf(u1a[r], 0.0f) * g1a[r];
            ps1[m * 32 + lo]      = fmaxf(u0b[r], 0.0f) * g0b[r];
            ps1[m * 32 + 16 + lo] = fmaxf(u1b[r], 0.0f) * g1b[r];
        }

        // h += p @ Wd^T : residual folded by seeding C with current h;
        // Wd fragment shared by both tiles.
        #pragma unroll
        for (int kk = 0; kk < 8; ++kk) {
            const v2f bD  = *(const v2f*)&wdS[lo * 32 + 4 * kk + 2 * hi];
            const v2f aP0 = *(const v2f*)&ps0[lo * 32 + 4 * kk + 2 * hi];
            const v2f aP1 = *(const v2f*)&ps1[lo * 32 + 4 * kk + 2 * hi];
            h0 = __builtin_amdgcn_wmma_f32_16x16x4_f32(false, aP0, false, bD, (short)0, h0, false, false);
            h1 = __builtin_amdgcn_wmma_f32_16x16x4_f32(false, aP1, false, bD, (short)0, h1, false, false);
        }

        // Publish new h tiles for the next block's A-fragment gathers.
        #pragma unroll
        for (int r = 0; r < 8; ++r) {
            const int m = r + 8 * hi;
            hs0[m * 16 + lo] = h0[r];
            hs1[m * 16 + lo] = h1[r];
        }
    };

    // ---- load both h tiles into C/D-layout registers, mirror into LDS ----
    v8f h0, h1;
    #pragma unroll
    for (int r = 0; r < 8; ++r) {
        const int m = r + 8 * hi;
        h0[r] = x[(row0 + m) * 16 + lo];
        h1[r] = x[(row1 + m) * 16 + lo];
    }
    #pragma unroll
    for (int r = 0; r < 8; ++r) {
        const int m = r + 8 * hi;
        hs0[m * 16 + lo] = h0[r];
        hs1[m * 16 + lo] = h1[r];
    }

    stage_group(0, 0);
    wait_async0();
    __syncthreads();

    for (int base = 0; base < n_blk; base += SB) {
        const int gbuf = (base / SB) & 1;
        if (base + SB < n_blk) stage_group(gbuf ^ 1, base + SB);  // overlap

        // Fully unrolled epoch: compile-time slab offsets fold into DS
        // immediates; tail guard is workgroup-uniform (no divergent barriers).
        do_block(&WuS[gbuf][0],   &WgS[gbuf][0],   &WdS[gbuf][0],   &bS[gbuf][0],  h0, h1);
        if (base + 1 < n_blk)
            do_block(&WuS[gbuf][512], &WgS[gbuf][512], &WdS[gbuf][512], &bS[gbuf][64], h0, h1);

        // Next epoch's weights must have landed; the barrier also fences
        // group-buffer reuse across the workgroup.
        wait_async0();
        __syncthreads();
    }

    #pragma unroll
    for (int r = 0; r < 8; ++r) {
        const int m = r + 8 * hi;
        out[(row0 + m) * 16 + lo] = h0[r];
        out[(row1 + m) * 16 + lo] = h1[r];
    }
}

extern "C" void kernel_launch(void* const* d_in, const int* in_sizes, int n_in,
                              void* d_out, int out_size, void* d_ws, size_t ws_size,
                              hipStream_t stream) {
    const float* x  = (const float*)d_in[0];
    const float* Wu = (const float*)d_in[1];
    const float* bu = (const float*)d_in[2];
    const float* Wg = (const float*)d_in[3];
    const float* bg = (const float*)d_in[4];
    const float* Wd = (const float*)d_in[5];
    float* out = (float*)d_out;

    const int n_rows = in_sizes[0] / 16;          // 262144
    const int n_blk  = in_sizes[1] / (32 * 16);   // 195
    const int ntiles = n_rows / 16;
    const int tiles_per_wg = WAVES_PER_WG * TPW;
    const int grid = (ntiles + tiles_per_wg - 1) / tiles_per_wg;

    palu_chain_kernel<<<grid, THREADS, 0, stream>>>(
        x, Wu, bu, Wg, bg, Wd, out, n_rows, n_blk);
}